// ConfidenceGuidedGate_82231443849381
// MI455X (gfx1250) — compile-verified
//
#include <hip/hip_runtime.h>
#include <hip/hip_bf16.h>

typedef __attribute__((ext_vector_type(2))) float v2f;
typedef __attribute__((ext_vector_type(8))) float v8f;

#define N_ROWS 16384
#define DDIM   4096
#define EDIM   128
#define TOPK   2
#define KC     32          // K-chunk staged in LDS per iteration
#define MBLK   64          // rows per workgroup (4 sub-tiles of 16)
#define XS_STR 36          // padded LDS row stride (floats): 16B-aligned stores, conflict-free b64 frag reads
#define WS_STR 36
#define CS_STR 130         // logit staging stride: conflict-free row scans
#define NCHUNK (DDIM / KC) // 128

#if defined(__gfx1250__) && __has_builtin(__builtin_amdgcn_global_load_async_to_lds_b128)
#define HAS_ASYNC 1
#else
#define HAS_ASYNC 0
#endif

#if HAS_ASYNC
typedef int v4i_vec __attribute__((vector_size(16)));                 // matches builtin proto
typedef __attribute__((address_space(1))) v4i_vec gv4;                // global (as1) int4
typedef __attribute__((address_space(3))) v4i_vec lv4;                // LDS (as3) int4
#if __has_builtin(__builtin_amdgcn_s_wait_asynccnt)
#define WAIT_ASYNC(n) __builtin_amdgcn_s_wait_asynccnt(n)
#else
#define WAIT_ASYNC(n) asm volatile("s_wait_asynccnt %0" :: "n"(n))
#endif
#endif

__global__ __launch_bounds__(256) void
ConfidenceGuidedGate_gemm_topk(const float* __restrict__ x,
                               const float* __restrict__ W,
                               const float* __restrict__ bvec,
                               float* __restrict__ out_vals,
                               int*   __restrict__ out_idx)
{
#if HAS_ASYNC
    __shared__ float xs[3][MBLK * XS_STR];   // 27.6 KB  x tiles  (triple-buffered)
    __shared__ float ws[3][EDIM * WS_STR];   // 55.3 KB  W tiles  (triple-buffered)
#else
    __shared__ float xs[1][MBLK * XS_STR];
    __shared__ float ws[1][EDIM * WS_STR];
#endif
    __shared__ float cs[MBLK * CS_STR];      // 33.3 KB  logits  (64 x 128)

    const int tid  = threadIdx.x;
    const int lane = tid & 31;
    const int wave = tid >> 5;            // 0..7 -> owns columns wave*16..+15
    const int lr   = lane & 15;
    const int half = lane >> 4;           // 0: K=k0,k0+1   1: K=k0+2,k0+3
    const int row0 = blockIdx.x * MBLK;

    // ---- cooperative staging mapping (all 128B-coalesced) ----
    const int xrow = tid >> 2;            // 0..63
    const int xk   = (tid & 3) * 8;       // 8 floats per thread
    const int we   = tid >> 1;            // 0..127
    const int wk   = (tid & 1) * 16;      // 16 floats per thread

    const float* xg = x + (size_t)(row0 + xrow) * DDIM + xk;
    const float* wg = W + (size_t)we * DDIM + wk;

    v8f acc0 = {}, acc1 = {}, acc2 = {}, acc3 = {};

#if HAS_ASYNC
    // ---- async global -> LDS staging (no VGPR round-trip, one barrier/chunk) ----
    auto issue = [&](int c, int buf) {
        const float* xp = xg + c * KC;
        const float* wp = wg + c * KC;
        float* xd = &xs[buf][xrow * XS_STR + xk];
        float* wd = &ws[buf][we   * WS_STR + wk];
        __builtin_amdgcn_global_load_async_to_lds_b128((gv4*)(xp + 0),  (lv4*)(xd + 0),  0, 0);
        __builtin_amdgcn_global_load_async_to_lds_b128((gv4*)(xp + 4),  (lv4*)(xd + 4),  0, 0);
        __builtin_amdgcn_global_load_async_to_lds_b128((gv4*)(wp + 0),  (lv4*)(wd + 0),  0, 0);
        __builtin_amdgcn_global_load_async_to_lds_b128((gv4*)(wp + 4),  (lv4*)(wd + 4),  0, 0);
        __builtin_amdgcn_global_load_async_to_lds_b128((gv4*)(wp + 8),  (lv4*)(wd + 8),  0, 0);
        __builtin_amdgcn_global_load_async_to_lds_b128((gv4*)(wp + 12), (lv4*)(wd + 12), 0, 0);
    };

    issue(0, 0);
    issue(1, 1);

    for (int c = 0; c < NCHUNK; ++c) {
        // keep two chunks in flight; async loads retire in order, so asynccnt<=12
        // guarantees chunk c (the oldest 6 ops) has landed in LDS.
        if (c + 2 < NCHUNK)      { issue(c + 2, (c + 2) % 3); WAIT_ASYNC(12); }
        else if (c + 1 < NCHUNK) { WAIT_ASYNC(6); }
        else                     { WAIT_ASYNC(0); }
        __syncthreads();          // single barrier per chunk (triple buffering)

        const float* xb = xs[c % 3];
        const float* wb = ws[c % 3];
        #pragma unroll
        for (int ks = 0; ks < KC / 4; ++ks) {
            const int k0 = ks * 4 + 2 * half;  // even -> 8B aligned
            v2f bfrag = *(const v2f*)&wb[(wave * 16 + lr) * WS_STR + k0];

            v2f a0 = *(const v2f*)&xb[( 0 + lr) * XS_STR + k0];
            v2f a1 = *(const v2f*)&xb[(16 + lr) * XS_STR + k0];
            v2f a2 = *(const v2f*)&xb[(32 + lr) * XS_STR + k0];
            v2f a3 = *(const v2f*)&xb[(48 + lr) * XS_STR + k0];

            acc0 = __builtin_amdgcn_wmma_f32_16x16x4_f32(false, a0, false, bfrag, (short)0, acc0, false, false);
            acc1 = __builtin_amdgcn_wmma_f32_16x16x4_f32(false, a1, false, bfrag, (short)0, acc1, false, false);
            acc2 = __builtin_amdgcn_wmma_f32_16x16x4_f32(false, a2, false, bfrag, (short)0, acc2, false, false);
            acc3 = __builtin_amdgcn_wmma_f32_16x16x4_f32(false, a3, false, bfrag, (short)0, acc3, false, false);
        }
    }
    __syncthreads();
#else
    // ---- fallback: register-prefetch staging (proven path) ----
    float4 xr0 = *(const float4*)(xg + 0);
    float4 xr1 = *(const float4*)(xg + 4);
    float4 wr0 = *(const float4*)(wg + 0);
    float4 wr1 = *(const float4*)(wg + 4);
    float4 wr2 = *(const float4*)(wg + 8);
    float4 wr3 = *(const float4*)(wg + 12);

    for (int c = 0; c < NCHUNK; ++c) {
        *(float4*)&xs[0][xrow * XS_STR + xk + 0]  = xr0;
        *(float4*)&xs[0][xrow * XS_STR + xk + 4]  = xr1;
        *(float4*)&ws[0][we   * WS_STR + wk + 0]  = wr0;
        *(float4*)&ws[0][we   * WS_STR + wk + 4]  = wr1;
        *(float4*)&ws[0][we   * WS_STR + wk + 8]  = wr2;
        *(float4*)&ws[0][we   * WS_STR + wk + 12] = wr3;
        __syncthreads();

        if (c + 1 < NCHUNK) {
            xg += KC; wg += KC;
            xr0 = *(const float4*)(xg + 0);
            xr1 = *(const float4*)(xg + 4);
            wr0 = *(const float4*)(wg + 0);
            wr1 = *(const float4*)(wg + 4);
            wr2 = *(const float4*)(wg + 8);
            wr3 = *(const float4*)(wg + 12);
        }

        #pragma unroll
        for (int ks = 0; ks < KC / 4; ++ks) {
            const int k0 = ks * 4 + 2 * half;
            v2f bfrag = *(const v2f*)&ws[0][(wave * 16 + lr) * WS_STR + k0];

            v2f a0 = *(const v2f*)&xs[0][( 0 + lr) * XS_STR + k0];
            v2f a1 = *(const v2f*)&xs[0][(16 + lr) * XS_STR + k0];
            v2f a2 = *(const v2f*)&xs[0][(32 + lr) * XS_STR + k0];
            v2f a3 = *(const v2f*)&xs[0][(48 + lr) * XS_STR + k0];

            acc0 = __builtin_amdgcn_wmma_f32_16x16x4_f32(false, a0, false, bfrag, (short)0, acc0, false, false);
            acc1 = __builtin_amdgcn_wmma_f32_16x16x4_f32(false, a1, false, bfrag, (short)0, acc1, false, false);
            acc2 = __builtin_amdgcn_wmma_f32_16x16x4_f32(false, a2, false, bfrag, (short)0, acc2, false, false);
            acc3 = __builtin_amdgcn_wmma_f32_16x16x4_f32(false, a3, false, bfrag, (short)0, acc3, false, false);
        }
        __syncthreads();
    }
#endif

    // ---- stage logits (+bias) to LDS; C layout: vgpr j, lanes 0-15 -> M=j, lanes 16-31 -> M=j+8 ----
    const float bias = bvec[wave * 16 + lr];
    const int   col  = wave * 16 + lr;
    #pragma unroll
    for (int j = 0; j < 8; ++j) {
        cs[( 0 + j + 8 * half) * CS_STR + col] = acc0[j] + bias;
        cs[(16 + j + 8 * half) * CS_STR + col] = acc1[j] + bias;
        cs[(32 + j + 8 * half) * CS_STR + col] = acc2[j] + bias;
        cs[(48 + j + 8 * half) * CS_STR + col] = acc3[j] + bias;
    }
    __syncthreads();

    // ---- top-2 over logits (sigmoid is monotonic); strict '>' matches jax tie-break (lowest index) ----
    if (tid < MBLK) {
        const float* rowp = &cs[tid * CS_STR];
        float b1 = -__builtin_inff(), b2 = -__builtin_inff();
        int   i1 = 0, i2 = 0;
        #pragma unroll 8
        for (int j = 0; j < EDIM; ++j) {
            float v = rowp[j];
            if (v > b1)      { b2 = b1; i2 = i1; b1 = v; i1 = j; }
            else if (v > b2) { b2 = v;  i2 = j; }
        }
        const int g = row0 + tid;
        out_vals[g * 2 + 0] = 1.0f / (1.0f + __expf(-b1));
        out_vals[g * 2 + 1] = 1.0f / (1.0f + __expf(-b2));
        out_idx [g * 2 + 0] = i1;
        out_idx [g * 2 + 1] = i2;
    }
}

extern "C" void kernel_launch(void* const* d_in, const int* in_sizes, int n_in,
                              void* d_out, int out_size, void* d_ws, size_t ws_size,
                              hipStream_t stream) {
    const float* x = (const float*)d_in[0];
    const float* W = (const float*)d_in[1];
    const float* b = (const float*)d_in[2];

    float* out_vals = (float*)d_out;                              // [N, 2] f32
    int*   out_idx  = (int*)((float*)d_out + (size_t)N_ROWS * 2); // [N, 2] i32 (bit pattern)

    dim3 grid(N_ROWS / MBLK);  // 256 blocks
    dim3 block(256);           // 8 waves
    ConfidenceGuidedGate_gemm_topk<<<grid, block, 0, stream>>>(x, W, b, out_vals, out_idx);
}